// Yolov3LossNew_23252952941303
// MI455X (gfx1250) — compile-verified
//
#include <hip/hip_runtime.h>
#include <math.h>

// Problem constants (from reference)
#define NB 16
#define NA 3
#define NC 80
#define NH 64
#define NW 64
#define NT 512
#define NCELL (NB * NA * NH * NW)      // 196608
#define CH_STRIDE (NH * NW)            // 4096 floats between channels
#define THRESH 0.5f
#define OBJ_SCALE 1.0f
#define NOOBJ_SCALE 0.5f
#define RBLOCK 256
#define NPART (NCELL / RBLOCK)         // 768, multiple of 64

typedef float v2f __attribute__((ext_vector_type(2)));
typedef float v8f __attribute__((ext_vector_type(8)));

__device__ __forceinline__ float sigm(float z) { return 1.0f / (1.0f + expf(-z)); }
__device__ __forceinline__ float clampp(float p) {
  return fminf(fmaxf(p, 1e-12f), 1.0f - 1e-12f);
}

// ---------------------------------------------------------------- init masks
__global__ void k_init(float* __restrict__ noobj, int* __restrict__ cellinfo) {
  int i = blockIdx.x * blockDim.x + threadIdx.x;
  if (i < NCELL) { noobj[i] = 1.0f; cellinfo[i] = -1; }
}

// ------------------------------------------------------- per-target scatter
__global__ void k_targets(const float* __restrict__ target,
                          float* __restrict__ noobj, int* __restrict__ cellinfo,
                          float* __restrict__ tx, float* __restrict__ ty,
                          float* __restrict__ tw, float* __restrict__ th) {
  int t = blockIdx.x * blockDim.x + threadIdx.x;
  if (t >= NT) return;
  // anchors / STRIDE
  const float aw[3] = {1.25f, 2.0f, 4.125f};
  const float ah[3] = {1.625f, 3.75f, 2.875f};
  float bf  = target[t * 6 + 0];
  float lf  = target[t * 6 + 1];
  float gx  = target[t * 6 + 2] * (float)NW;
  float gy  = target[t * 6 + 3] * (float)NH;
  float gw  = target[t * 6 + 4] * (float)NW;
  float gh  = target[t * 6 + 5] * (float)NH;
  int b = (int)bf, lab = (int)lf;
  int gi = (int)gx, gj = (int)gy;

  float iou[3];
  float best = -1.0f; int bn = 0;
  for (int a = 0; a < 3; ++a) {
    float inter = fminf(aw[a], gw) * fminf(ah[a], gh);
    float uni   = aw[a] * ah[a] + 1e-16f + gw * gh - inter;
    iou[a] = inter / uni;
    if (iou[a] > best) { best = iou[a]; bn = a; }   // first max wins, like argmax
  }
  int cidx = ((b * NA + bn) * NH + gj) * NW + gi;
  cellinfo[cidx] = lab;
  tx[cidx] = gx - floorf(gx);
  ty[cidx] = gy - floorf(gy);
  tw[cidx] = logf(gw / aw[bn] + 1e-16f);
  th[cidx] = logf(gh / ah[bn] + 1e-16f);
  noobj[cidx] = 0.0f;                                // obj cell -> noobj 0
  for (int a = 0; a < 3; ++a)                        // keep==0 -> zero-store
    if (iou[a] > THRESH)
      noobj[((b * NA + a) * NH + gj) * NW + gi] = 0.0f;
}

// -------------------------------------------------- per-cell loss reduction
__global__ void k_reduce(const float* __restrict__ out,
                         const float* __restrict__ noobj,
                         const int* __restrict__ cellinfo,
                         const float* __restrict__ tx, const float* __restrict__ ty,
                         const float* __restrict__ tw, const float* __restrict__ th,
                         float* __restrict__ partials) {
  int idx = blockIdx.x * blockDim.x + threadIdx.x;   // cell index
  int i  = idx & (NW - 1);
  int j  = (idx >> 6) & (NH - 1);
  int ab = idx >> 12;                                // b*NA + a  (channel block)
  // layout (NB, NA*85, NH, NW): channel (a*85+c) -> offset (ab*85 + c)*4096
  const float* p = out + (size_t)ab * 85u * CH_STRIDE + (size_t)j * NW + i;

  int   info = cellinfo[idx];
  float nm   = noobj[idx];
  float acc  = 0.0f;

  // conf channel: needed by obj and noobj terms
  float pc = clampp(sigm(p[4 * CH_STRIDE]));
  if (info >= 0) acc += OBJ_SCALE * (-logf(pc));           // tconf=1 at obj cells
  acc += NOOBJ_SCALE * nm * (-logf(1.0f - pc));            // tconf=0 where nm>0

  if (info >= 0) {                                         // sparse: <=512 cells
    float dx = sigm(p[0])             - tx[idx]; acc += dx * dx;
    float dy = sigm(p[CH_STRIDE])     - ty[idx]; acc += dy * dy;
    float dw = p[2 * CH_STRIDE]       - tw[idx]; acc += dw * dw;
    float dh = p[3 * CH_STRIDE]       - th[idx]; acc += dh * dh;
    for (int c = 0; c < NC; ++c) {
      float pcl = clampp(sigm(p[(5 + c) * CH_STRIDE]));
      acc += (c == info) ? -logf(pcl) : -logf(1.0f - pcl);
    }
  }

  // wave32 reduction
  for (int off = 16; off > 0; off >>= 1) acc += __shfl_xor(acc, off, 32);
  __shared__ float wsum[RBLOCK / 32];
  int lane = threadIdx.x & 31, wid = threadIdx.x >> 5;
  if (lane == 0) wsum[wid] = acc;
  __syncthreads();
  if (threadIdx.x == 0) {
    float s = 0.0f;
    for (int w = 0; w < RBLOCK / 32; ++w) s += wsum[w];
    partials[blockIdx.x] = s;
  }
}

// ---------------------- final reduction: WMMA f32 16x16x4 (D = A x ones + C)
// With B == ones, every column n of D accumulates sum_{m,k} A[m][k]; summing a
// column (lane n regs 0..7  +  lane n+16 regs 0..7) yields the grand total in
// full fp32 precision, deterministically. One wave, EXEC all ones.
__global__ void k_final(const float* __restrict__ partials, float* __restrict__ d_out) {
  int lane = threadIdx.x;     // 0..31
#if __has_builtin(__builtin_amdgcn_wmma_f32_16x16x4_f32)
  v8f c = {};
  v2f ones; ones.x = 1.0f; ones.y = 1.0f;
  for (int base = 0; base < NPART; base += 64) {       // 768 = 12 * 64, no mask
    v2f a;
    a.x = partials[base + lane];
    a.y = partials[base + 32 + lane];
    c = __builtin_amdgcn_wmma_f32_16x16x4_f32(
        /*neg_a=*/false, a, /*neg_b=*/false, ones,
        /*c_mod=*/(short)0, c, /*reuse_a=*/false, /*reuse_b=*/false);
  }
  float s = c[0] + c[1] + c[2] + c[3] + c[4] + c[5] + c[6] + c[7];
  s += __shfl_xor(s, 16, 32);                           // column n + column n (M=8..15)
  if (lane == 0) d_out[0] = s / (float)NB;
#else
  float s = 0.0f;
  for (int g = lane; g < NPART; g += 32) s += partials[g];
  for (int off = 16; off > 0; off >>= 1) s += __shfl_xor(s, off, 32);
  if (lane == 0) d_out[0] = s / (float)NB;
#endif
}

// --------------------------------------------------------------------- host
extern "C" void kernel_launch(void* const* d_in, const int* in_sizes, int n_in,
                              void* d_out, int out_size, void* d_ws, size_t ws_size,
                              hipStream_t stream) {
  const float* output = (const float*)d_in[0];   // (NB, NA*85, NH, NW) fp32
  const float* target = (const float*)d_in[1];   // (NT, 6) fp32

  float* ws       = (float*)d_ws;
  float* noobj    = ws;                    // NCELL
  float* tx       = noobj + NCELL;         // NCELL
  float* ty       = tx + NCELL;
  float* tw       = ty + NCELL;
  float* th       = tw + NCELL;
  float* partials = th + NCELL;            // NPART
  int*   cellinfo = (int*)(partials + NPART); // NCELL ints
  (void)in_sizes; (void)n_in; (void)out_size; (void)ws_size;

  k_init   <<<NCELL / 256, 256, 0, stream>>>(noobj, cellinfo);
  k_targets<<<(NT + 255) / 256, 256, 0, stream>>>(target, noobj, cellinfo, tx, ty, tw, th);
  k_reduce <<<NCELL / RBLOCK, RBLOCK, 0, stream>>>(output, noobj, cellinfo,
                                                   tx, ty, tw, th, partials);
  k_final  <<<1, 32, 0, stream>>>(partials, (float*)d_out);
}